// TestModel_89060441850633
// MI455X (gfx1250) — compile-verified
//
#include <hip/hip_runtime.h>

// Problem constants (match reference)
#define TT 4096
#define BB 64
#define HH 4
#define KPP 15
#define TB (TT*BB)
#define NTILE (TB/16)   // 16-row tiles of the flattened (T*B) dim

typedef __attribute__((ext_vector_type(2))) float v2f;
typedef __attribute__((ext_vector_type(8))) float v8f;

// D(16x16,f32) = A(16x4,f32) x B(4x16,f32) + C(16x16,f32)
static __device__ __forceinline__ v8f wmma4(v2f a, v2f b, v8f c) {
  return __builtin_amdgcn_wmma_f32_16x16x4_f32(false, a, false, b, (short)0, c,
                                               false, false);
}

static __device__ __forceinline__ float fsig(float x) {
  return __builtin_amdgcn_rcpf(1.0f + __expf(-x));
}
static __device__ __forceinline__ float ftanh_(float x) {
  // tanh(x) = 1 - 2/(e^{2x}+1); saturates correctly at +-inf
  return 1.0f - 2.0f * __builtin_amdgcn_rcpf(__expf(2.0f * x) + 1.0f);
}

// Branchless guarded load: unconditional load at clamped index + cndmask.
// Avoids s_and_saveexec/s_cbranch around scalar loads (divergent predicates).
static __device__ __forceinline__ float ldm(const float* __restrict__ p,
                                            int idx, int idx_max, bool valid) {
  int ci = idx < idx_max ? idx : idx_max;
  float v = p[ci];
  return valid ? v : 0.0f;
}

// ---------------------------------------------------------------------------
// Phase 0 (parallel): propri_ft = propri@enc_W.T + enc_b (kept internal),
//   xwc = propri_ft@core_Wih.T + core_bih + core_bhh
//   xwp = propri_ft@Wih[:,4:8].T + propri@Wih[:,8:23].T + biases
// One wave per 16-row tile. Results stored LANE-MAJOR per tile:
//   xw[(tile*32 + lane)*8 + j]  (j = C-register index)
// so the scan reloads each tile as two b128 loads per lane.
// ---------------------------------------------------------------------------
__global__ void __launch_bounds__(256)
precompute_kernel(const float* __restrict__ propri,
                  const float* __restrict__ enc_W, const float* __restrict__ enc_b,
                  const float* __restrict__ core_Wih,
                  const float* __restrict__ core_bih, const float* __restrict__ core_bhh,
                  const float* __restrict__ pol_Wih,
                  const float* __restrict__ pol_bih, const float* __restrict__ pol_bhh,
                  float* __restrict__ xwc, float* __restrict__ xwp)
{
  __shared__ float lds[8][16 * 17];
  const int lane = threadIdx.x & 31;
  const int wave = threadIdx.x >> 5;
  const int tile = blockIdx.x * 8 + wave;
  const int R0   = tile << 4;        // 16 consecutive flattened rows
  const int m    = lane & 15;        // A row / C column index
  const int half = lane >> 4;
  const int k0   = half << 1;        // K slot base for A/B operands
  const int n    = m;                // B/C column (gate index)

  // A operand chunks of propri, K padded 15 -> 16 (4 chunks of K=4)
  v2f a[4];
#pragma unroll
  for (int c = 0; c < 4; ++c) {
#pragma unroll
    for (int j = 0; j < 2; ++j) {
      int kk = 4 * c + k0 + j;
      a[c][j] = ldm(propri + (size_t)(R0 + m) * KPP, kk, KPP - 1, kk < KPP);
    }
  }

  // Encoder: propri_ft tile in C layout (columns >= 4 are exactly zero)
  const bool ncol = (n < HH);
  const int  ncl  = ncol ? n : (HH - 1);   // clamped row of enc_W / bias
  v8f ft;
  {
    float eb = enc_b[ncl];
#pragma unroll
    for (int j = 0; j < 8; ++j) ft[j] = ncol ? eb : 0.0f;
  }
#pragma unroll
  for (int c = 0; c < 4; ++c) {
    v2f bv;
#pragma unroll
    for (int j = 0; j < 2; ++j) {
      int kk = 4 * c + k0 + j;
      bv[j] = ldm(enc_W + ncl * KPP, kk, KPP - 1, ncol && (kk < KPP));
    }
    ft = wmma4(a[c], bv, ft);
  }

  // Transpose ft (C layout) -> A layout through padded LDS
  float* L = lds[wave];
#pragma unroll
  for (int j = 0; j < 8; ++j) L[(j + 8 * half) * 17 + n] = ft[j];
  __builtin_amdgcn_wave_barrier();
  v2f aft;
#pragma unroll
  for (int j = 0; j < 2; ++j) aft[j] = L[m * 17 + k0 + j];

  // xw_core = ft @ core_Wih.T + (bih + bhh): one exact 16x16x4 WMMA
  v8f acc;
  {
    float bias = core_bih[n] + core_bhh[n];
#pragma unroll
    for (int j = 0; j < 8; ++j) acc[j] = bias;
  }
  v2f bw;
#pragma unroll
  for (int j = 0; j < 2; ++j) bw[j] = core_Wih[n * HH + k0 + j];
  acc = wmma4(aft, bw, acc);
  *(v8f*)(xwc + ((size_t)tile * 32 + lane) * 8) = acc;   // 2x b128 store

  // xw_pol partial: ft part (cols 4..7) + propri part (cols 8..22)
  v8f accp;
  {
    float bias = pol_bih[n] + pol_bhh[n];
#pragma unroll
    for (int j = 0; j < 8; ++j) accp[j] = bias;
  }
  v2f bp;
#pragma unroll
  for (int j = 0; j < 2; ++j) bp[j] = pol_Wih[n * 23 + 4 + k0 + j];
  accp = wmma4(aft, bp, accp);
#pragma unroll
  for (int c = 0; c < 4; ++c) {
    v2f bq;
#pragma unroll
    for (int j = 0; j < 2; ++j) {
      int col = 8 + 4 * c + k0 + j;
      bq[j] = ldm(pol_Wih + n * 23, col, 22, col < 23);
    }
    accp = wmma4(a[c], bq, accp);
  }
  *(v8f*)(xwp + ((size_t)tile * 32 + lane) * 8) = accp;  // 2x b128 store
}

// ---------------------------------------------------------------------------
// Phase 1 (serial): fused dual-LSTM scan with per-step resets.
// 4 independent single-wave workgroups; wave w owns batches 16w..16w+15.
// State h,c live in VGPRs in WMMA A-matrix layout across all 4096 steps.
// ---------------------------------------------------------------------------
__global__ void __launch_bounds__(32)
scan_kernel(const float* __restrict__ xwc, const float* __restrict__ xwp,
            const int* __restrict__ done,
            const float* __restrict__ core_Whh,
            const float* __restrict__ pol_Wih, const float* __restrict__ pol_Whh,
            float* __restrict__ core_out, float* __restrict__ pol_out)
{
  __shared__ float lds[16 * 17];
  const int lane = threadIdx.x & 31;
  const int blk  = blockIdx.x;       // 0..3 -> batch group
  const int m    = lane & 15;
  const int half = lane >> 4;
  const int k0   = half << 1;
  const int n    = m;
  const int b    = blk * 16 + m;     // this lane's batch (A-row)

  // Recurrent B operands, resident for the whole scan
  v2f Bc, Bp0, Bph;
#pragma unroll
  for (int j = 0; j < 2; ++j) {
    Bc[j]  = core_Whh[n * HH + k0 + j];     // core h @ Whh.T
    Bp0[j] = pol_Wih[n * 23 + k0 + j];      // policy: core_out contribution
    Bph[j] = pol_Whh[n * HH + k0 + j];      // policy h @ Whh.T
  }

  v2f hC = {0.0f, 0.0f}, cC = hC, hP = hC, cP = hC;

  for (int t = 0; t < TT; ++t) {
    const int rowbase = t * BB + blk * 16;
    const int tile    = t * 4 + blk;            // lane-major tile index

    // reset[t][b] = (t==0) | done[b][t-1]  (branchless)
    float keep = (t == 0) ? 0.0f
                          : (done[(size_t)b * TT + (t - 1)] ? 0.0f : 1.0f);
    hC *= keep; cC *= keep; hP *= keep; cP *= keep;

    // Precomputed input projections: lane-major -> 2x b128 loads each
    v8f Cc = *(const v8f*)(xwc + ((size_t)tile * 32 + lane) * 8);
    v8f Cp = *(const v8f*)(xwp + ((size_t)tile * 32 + lane) * 8);

    // Prefetch next step's tiles (1 KB contiguous each)
    if (t + 1 < TT) {
      const char* p0 = (const char*)(xwc + ((size_t)(tile + 4) * 32) * 8) + lane * 32;
      const char* p1 = (const char*)(xwp + ((size_t)(tile + 4) * 32) * 8) + lane * 32;
      __builtin_prefetch(p0, 0, 3);
      __builtin_prefetch(p1, 0, 3);
    }

    // ---- core LSTM step: gates = h @ Whh.T + xw  (one 16x16x4 WMMA)
    v8f Dc = wmma4(hC, Bc, Cc);
#pragma unroll
    for (int j = 0; j < 8; ++j) lds[(j + 8 * half) * 17 + n] = Dc[j];
    __builtin_amdgcn_wave_barrier();
#pragma unroll
    for (int j = 0; j < 2; ++j) {            // gate order i,f,g,o
      int hid = k0 + j;
      float iv = lds[m * 17 + hid];
      float fv = lds[m * 17 + 4 + hid];
      float gv = lds[m * 17 + 8 + hid];
      float ov = lds[m * 17 + 12 + hid];
      float cn = fsig(fv) * cC[j] + fsig(iv) * ftanh_(gv);
      cC[j] = cn;
      hC[j] = fsig(ov) * ftanh_(cn);
    }
    __builtin_amdgcn_wave_barrier();
    // this lane's two hidden values are adjacent in memory -> one b64 store
    *(v2f*)(core_out + (size_t)(rowbase + m) * HH + k0) = hC;

    // ---- policy LSTM step: gates = xw + core_out@W0.T + h@Whh.T (chained)
    v8f Dp = wmma4(hC, Bp0, Cp);
    Dp = wmma4(hP, Bph, Dp);
#pragma unroll
    for (int j = 0; j < 8; ++j) lds[(j + 8 * half) * 17 + n] = Dp[j];
    __builtin_amdgcn_wave_barrier();
#pragma unroll
    for (int j = 0; j < 2; ++j) {
      int hid = k0 + j;
      float iv = lds[m * 17 + hid];
      float fv = lds[m * 17 + 4 + hid];
      float gv = lds[m * 17 + 8 + hid];
      float ov = lds[m * 17 + 12 + hid];
      float cn = fsig(fv) * cP[j] + fsig(iv) * ftanh_(gv);
      cP[j] = cn;
      hP[j] = fsig(ov) * ftanh_(cn);
    }
    __builtin_amdgcn_wave_barrier();
    *(v2f*)(pol_out + (size_t)(rowbase + m) * HH + k0) = hP;
  }
}

// ---------------------------------------------------------------------------
// Phase 2 (parallel epilogue): 5-layer 4x4 ReLU MLP + 4->1 head on core_out.
// ---------------------------------------------------------------------------
typedef __attribute__((ext_vector_type(4))) float v4f;

__global__ void __launch_bounds__(256)
value_kernel(const float* __restrict__ core_out,
             const float* __restrict__ mlp_W, const float* __restrict__ mlp_b,
             const float* __restrict__ out_W, const float* __restrict__ out_b,
             float* __restrict__ val_out)
{
  int row = blockIdx.x * blockDim.x + threadIdx.x;
  if (row >= TB) return;
  v4f hv = *(const v4f*)(core_out + (size_t)row * 4);  // one b128 load
  float h[4] = {hv[0], hv[1], hv[2], hv[3]};
#pragma unroll
  for (int l = 0; l < 5; ++l) {
    float nh[4];
#pragma unroll
    for (int j = 0; j < 4; ++j) {
      float s = mlp_b[l * 4 + j];
#pragma unroll
      for (int k = 0; k < 4; ++k) s += mlp_W[(l * 4 + j) * 4 + k] * h[k];
      nh[j] = fmaxf(s, 0.0f);
    }
#pragma unroll
    for (int j = 0; j < 4; ++j) h[j] = nh[j];
  }
  float v = out_b[0];
#pragma unroll
  for (int k = 0; k < 4; ++k) v += out_W[k] * h[k];
  val_out[row] = v;
}

extern "C" void kernel_launch(void* const* d_in, const int* in_sizes, int n_in,
                              void* d_out, int out_size, void* d_ws, size_t ws_size,
                              hipStream_t stream)
{
  // setup_inputs order: 0 visual (unused), 1 propri, 2 done, 3 enc_W, 4 enc_b,
  // 5 core_Wih, 6 core_Whh, 7 core_bih, 8 core_bhh, 9 pol_Wih, 10 pol_Whh,
  // 11 pol_bih, 12 pol_bhh, 13 mlp_W, 14 mlp_b, 15 out_W, 16 out_b
  const float* propri   = (const float*)d_in[1];
  const int*   done     = (const int*)d_in[2];
  const float* enc_W    = (const float*)d_in[3];
  const float* enc_b    = (const float*)d_in[4];
  const float* core_Wih = (const float*)d_in[5];
  const float* core_Whh = (const float*)d_in[6];
  const float* core_bih = (const float*)d_in[7];
  const float* core_bhh = (const float*)d_in[8];
  const float* pol_Wih  = (const float*)d_in[9];
  const float* pol_Whh  = (const float*)d_in[10];
  const float* pol_bih  = (const float*)d_in[11];
  const float* pol_bhh  = (const float*)d_in[12];
  const float* mlp_W    = (const float*)d_in[13];
  const float* mlp_b    = (const float*)d_in[14];
  const float* out_W    = (const float*)d_in[15];
  const float* out_b    = (const float*)d_in[16];

  float* ws   = (float*)d_ws;
  float* xwc  = ws;                          // (NTILE, 32 lanes, 8) lane-major
  float* xwp  = xwc + (size_t)TB * 16;       // same layout
  float* cout = xwp + (size_t)TB * 16;       // (T*B,4)

  float* val = (float*)d_out;                // (T*B,) value
  float* pol = val + (size_t)TB;             // (T*B,4) policy mean

  precompute_kernel<<<NTILE / 8, 256, 0, stream>>>(
      propri, enc_W, enc_b, core_Wih, core_bih, core_bhh,
      pol_Wih, pol_bih, pol_bhh, xwc, xwp);

  scan_kernel<<<4, 32, 0, stream>>>(
      xwc, xwp, done, core_Whh, pol_Wih, pol_Whh, cout, pol);

  value_kernel<<<(TB + 255) / 256, 256, 0, stream>>>(
      cout, mlp_W, mlp_b, out_W, out_b, val);
}